// RegionLoss_65755949301935
// MI455X (gfx1250) — compile-verified
//
#include <hip/hip_runtime.h>
#include <hip/hip_bf16.h>
#include <stdint.h>

// ---------------- problem constants ----------------
#define NA_     5
#define NC_     13
#define NT_     50
#define NH_     32
#define NW_     32
#define NB_     16
#define CH_     32                    // 19 + 13 channels per anchor
#define PLANE_  (NH_*NW_)             // 1024
#define M_      (NA_*PLANE_)          // 5120
#define NTOT_   (NB_*M_)              // 81920

#define TPAD_       64                // targets padded to 4 WMMA tiles of 16
#define GT_STRIDE_  20
#define CELL_STRIDE_ 24

#define SIL_THRESH_ 0.6f
#define OBJ_SCALE_  5.0f

// ---------------- workspace layout (floats) ----------------
#define NBLK_B_      640
#define WS_PARTIALS_ 0
#define WS_GTW_      (WS_PARTIALS_ + NBLK_B_)                  // 16*64*20 = 20480
#define WS_CELL_     (WS_GTW_ + NB_*TPAD_*GT_STRIDE_)          // 16*1024*24 = 393216

typedef float v2f __attribute__((ext_vector_type(2)));
typedef float v8f __attribute__((ext_vector_type(8)));

__device__ __forceinline__ float corner_conf_term(float d) {
    // (exp(2*(1 - d/80)) - 1) / (e^2 - 1), zero when d >= 80
    const float inv = 0.15651764274966565f;       // 1/(e^2-1)
    float e = __expf(2.0f - d * 0.025f);
    float c = (e - 1.0f) * inv;
    return (d < 80.0f) ? c : 0.0f;
}

__device__ __forceinline__ float sigmoidf_(float x) {
    return 1.0f / (1.0f + __expf(-x));
}

// ================= Kernel A: per-batch target preprocessing =================
// 16 blocks x 32 threads (1 wave per batch). Serial in t -> deterministic
// last-wins scatter, matching the reference .at[flat].set semantics.
__global__ void ka_targets(const float* __restrict__ out,
                           const float* __restrict__ target,
                           float* __restrict__ ws) {
    const int b    = blockIdx.x;
    const int lane = threadIdx.x;
    const float* tg = target + (size_t)b * (NT_ * 21);
    float* gtw   = ws + WS_GTW_  + (size_t)b * TPAD_ * GT_STRIDE_;
    float* cellT = ws + WS_CELL_ + (size_t)b * PLANE_ * CELL_STRIDE_;

    // zero targeted-flags for this batch
    for (int c = lane; c < PLANE_; c += 32) cellT[c * CELL_STRIDE_] = 0.0f;

    // nvalid = cumprod(tgt[:,1] != 0)  (uniform scan)
    int nvalid = NT_;
    for (int t = 0; t < NT_; ++t) {
        if (tg[t * 21 + 1] == 0.0f) { nvalid = t; break; }
    }

    // premultiplied (pixel-space) target corners; pad invalid with 1e9 -> dist>80 -> conf 0
    for (int idx = lane; idx < TPAD_ * GT_STRIDE_; idx += 32) {
        int t = idx / GT_STRIDE_;
        int e = idx % GT_STRIDE_;
        float v = 0.0f;
        if (e < 18) {
            if (t < nvalid) v = tg[t * 21 + 1 + e] * ((e & 1) ? 480.0f : 640.0f);
            else            v = 1.0e9f;
        }
        gtw[idx] = v;
    }
    __syncthreads();

    if (lane == 0) {
        for (int t = 0; t < nvalid; ++t) {
            float x0 = tg[t * 21 + 1];
            float y0 = tg[t * 21 + 2];
            int gi = (int)floorf(x0 * (float)NW_);
            int gj = (int)floorf(y0 * (float)NH_);
            if (gi < 0 || gi >= NW_ || gj < 0 || gj >= NH_) continue;   // mode='drop'
            int cell = gj * NW_ + gi;
            const float* p = out + (size_t)b * NA_ * CH_ * PLANE_ + (gj * NW_ + gi);
            float* ct = cellT + (size_t)cell * CELL_STRIDE_;
            float csum = 0.0f;
            #pragma unroll
            for (int c = 0; c < 9; ++c) {
                float gtx = tg[t * 21 + 1 + 2 * c];
                float gty = tg[t * 21 + 2 + 2 * c];
                float xsv = p[(2 * c) * PLANE_];
                float ysv = p[(2 * c + 1) * PLANE_];
                if (c == 0) { xsv = sigmoidf_(xsv); ysv = sigmoidf_(ysv); }
                float pcx = (xsv + (float)gi) * (1.0f / (float)NW_);
                float pcy = (ysv + (float)gj) * (1.0f / (float)NH_);
                float dx = (gtx - pcx) * 640.0f;
                float dy = (gty - pcy) * 480.0f;
                csum += corner_conf_term(sqrtf(dx * dx + dy * dy));
                ct[3 + 2 * c]     = gtx * (float)NW_ - (float)gi;        // tcoord x
                ct[3 + 2 * c + 1] = gty * (float)NH_ - (float)gj;        // tcoord y
            }
            ct[0] = 1.0f;                 // targeted flag (coord/cls mask)
            ct[1] = csum * (1.0f / 9.0f); // tconf
            ct[2] = tg[t * 21 + 0];       // tcls
        }
    }
}

// ================= Kernel B: main anchor-cell kernel =================
// 640 blocks x 256 threads; each wave owns 16 cells (both half-waves mirror
// the same 16 cells so WMMA A/B lane halves carry K0/K1 vs K2/K3).
__global__ void kb_main(const float* __restrict__ out,
                        float* __restrict__ ws) {
    __shared__ float sGt[TPAD_ * GT_STRIDE_];   // 1280 floats, per-batch target table
    __shared__ float sRed[256];

    const int tid  = threadIdx.x;
    const int lane = tid & 31;
    const int wid  = tid >> 5;

    const int n0 = blockIdx.x * 128;       // block never straddles a (b,a) plane
    const int b  = n0 / M_;                // uniform per block

    // ---- async stage: gtw[b] (global ws) -> LDS, tracked by ASYNCcnt ----
    const float* gsrc = ws + WS_GTW_ + (size_t)b * TPAD_ * GT_STRIDE_;
    #pragma unroll
    for (int it = 0; it < (TPAD_ * GT_STRIDE_) / 256; ++it) {
        int e = tid + it * 256;
        uint32_t ldsOff = (uint32_t)(uintptr_t)(&sGt[e]);   // low 32b of LDS flat addr
        uint32_t goff   = (uint32_t)(e * 4);
        asm volatile("global_load_async_to_lds_b32 %0, %1, %2"
                     :: "v"(ldsOff), "v"(goff), "s"(gsrc) : "memory");
    }
    asm volatile("s_wait_asynccnt 0" ::: "memory");
    __syncthreads();

    const int nidx = n0 + wid * 16 + (lane & 15);
    const int i = nidx & (NW_ - 1);
    const int j = (nidx >> 5) & (NH_ - 1);
    const int a = (nidx >> 10) % NA_;
    const float* p = out + (size_t)(b * NA_ + a) * CH_ * PLANE_ + (j * NW_ + i);

    float xs[9], ys[9], px[9], py[9];
    #pragma unroll
    for (int c = 0; c < 9; ++c) {
        xs[c] = p[(2 * c) * PLANE_];
        ys[c] = p[(2 * c + 1) * PLANE_];
    }
    xs[0] = sigmoidf_(xs[0]);
    ys[0] = sigmoidf_(ys[0]);
    float confv = sigmoidf_(p[18 * PLANE_]);
    #pragma unroll
    for (int c = 0; c < 9; ++c) {
        px[c] = (xs[c] + (float)i) * (640.0f / (float)NW_);  // pixel-space pred x
        py[c] = (ys[c] + (float)j) * (480.0f / (float)NH_);  // pixel-space pred y
    }

    const bool hiHalf = (lane >= 16);
    float curmax = 0.0f;

    // ---- 4 target tiles x 9 corners of V_WMMA_F32_16X16X4_F32 ----
    // A[t] = [gx, gy, gx^2+gy^2, 1], B[cell] = [-2px, -2py, 1, px^2+py^2]
    // => D[t,cell] = (gx-px)^2 + (gy-py)^2  (exact per-corner dist^2)
    for (int tt = 0; tt < TPAD_ / 16; ++tt) {
        float acc[8];
        #pragma unroll
        for (int v = 0; v < 8; ++v) acc[v] = 0.0f;
        #pragma unroll
        for (int c = 0; c < 9; ++c) {
            int t = tt * 16 + (lane & 15);
            float gx = sGt[t * GT_STRIDE_ + 2 * c];
            float gy = sGt[t * GT_STRIDE_ + 2 * c + 1];
            v2f A, B;
            A.x = hiHalf ? (gx * gx + gy * gy) : gx;       // K0 | K2
            A.y = hiHalf ? 1.0f : gy;                      // K1 | K3
            float pxc = px[c], pyc = py[c];
            B.x = hiHalf ? 1.0f : (-2.0f * pxc);           // K0 | K2
            B.y = hiHalf ? (pxc * pxc + pyc * pyc) : (-2.0f * pyc);  // K1 | K3
            v8f Cz = {};
            v8f D = __builtin_amdgcn_wmma_f32_16x16x4_f32(
                        false, A, false, B, (short)0, Cz, false, false);
            #pragma unroll
            for (int v = 0; v < 8; ++v) {
                float d = sqrtf(fmaxf(D[v], 0.0f));
                acc[v] += corner_conf_term(d);
            }
        }
        #pragma unroll
        for (int v = 0; v < 8; ++v)
            curmax = fmaxf(curmax, acc[v] * (1.0f / 9.0f));
    }
    // lane l holds targets {v+8*hi} for cell l%16 -> fold the two half-waves
    curmax = fmaxf(curmax, __shfl_xor(curmax, 16, 32));

    // ---- sparse tail (one lane per cell) ----
    float part = 0.0f;
    if (lane < 16) {
        float tconf = 0.0f;
        bool targeted = false;
        const float* ct = ws + WS_CELL_ +
            ((size_t)b * PLANE_ + (size_t)(j * NW_ + i)) * CELL_STRIDE_;
        if (a == 0 && ct[0] != 0.0f) { targeted = true; tconf = ct[1]; }

        float mask = targeted ? OBJ_SCALE_ : ((curmax > SIL_THRESH_) ? 0.0f : 1.0f);
        float dc = confv - tconf;
        part += 0.5f * mask * dc * dc;                      // loss_conf

        if (targeted) {
            #pragma unroll
            for (int c = 0; c < 9; ++c) {                   // loss_coord
                float ex = xs[c] - ct[3 + 2 * c];
                float ey = ys[c] - ct[3 + 2 * c + 1];
                part += 0.5f * (ex * ex + ey * ey);
            }
            float lg[13];                                   // loss_cls (CE)
            float mx = -1.0e30f;
            #pragma unroll
            for (int k = 0; k < NC_; ++k) {
                lg[k] = p[(19 + k) * PLANE_];
                mx = fmaxf(mx, lg[k]);
            }
            float se = 0.0f;
            #pragma unroll
            for (int k = 0; k < NC_; ++k) se += __expf(lg[k] - mx);
            float lse = __logf(se) + mx;
            int label = (int)ct[2];
            label = label < 0 ? 0 : (label > NC_ - 1 ? NC_ - 1 : label);
            part += lse - lg[label];
        }
    }

    // deterministic block reduction
    sRed[tid] = part;
    __syncthreads();
    for (int s = 128; s > 0; s >>= 1) {
        if (tid < s) sRed[tid] += sRed[tid + s];
        __syncthreads();
    }
    if (tid == 0) ws[WS_PARTIALS_ + blockIdx.x] = sRed[0];
}

// ================= Kernel C: final fixed-order reduction =================
__global__ void kc_reduce(const float* __restrict__ ws, float* __restrict__ outp) {
    __shared__ float s[256];
    int tid = threadIdx.x;
    float v = ws[WS_PARTIALS_ + tid] + ws[WS_PARTIALS_ + tid + 256];
    if (tid < 128) v += ws[WS_PARTIALS_ + tid + 512];
    s[tid] = v;
    __syncthreads();
    for (int st = 128; st > 0; st >>= 1) {
        if (tid < st) s[tid] += s[tid + st];
        __syncthreads();
    }
    if (tid == 0) outp[0] = s[0];
}

extern "C" void kernel_launch(void* const* d_in, const int* in_sizes, int n_in,
                              void* d_out, int out_size, void* d_ws, size_t ws_size,
                              hipStream_t stream) {
    const float* out    = (const float*)d_in[0];
    const float* target = (const float*)d_in[1];
    float* ws = (float*)d_ws;

    ka_targets<<<NB_, 32, 0, stream>>>(out, target, ws);
    kb_main<<<NBLK_B_, 256, 0, stream>>>(out, ws);
    kc_reduce<<<1, 256, 0, stream>>>(ws, (float*)d_out);
}